// AlignS_30442728194062
// MI455X (gfx1250) — compile-verified
//
#include <hip/hip_runtime.h>
#include <math.h>

#define BGR   64
#define NPER  1024
#define DIM   128
#define KDIM  64
#define EPER  16384

typedef __attribute__((ext_vector_type(2))) float v2f;
typedef __attribute__((ext_vector_type(8))) float v8f;

// ---------------------------------------------------------------------------
// ws layout (float offsets):
//   [0,        8388608)  agg  (B*N*D)    -- reused later as AS (B*N*K)
//   [8388608,  8454144)  deg  (B*N)
//   [8454144, 12648448)  S    (B*N*K)
//   [12648448]           ent_acc
//   [12648449]           l2_acc
// ---------------------------------------------------------------------------

// 1 wave per edge: agg[dst] += h[src] (float4/lane), deg[dst] += 1
__global__ void __launch_bounds__(256)
k_scatter_agg(const float* __restrict__ h, const int* __restrict__ src,
              const int* __restrict__ dst, float* __restrict__ agg,
              float* __restrict__ deg) {
    int wave = (blockIdx.x * blockDim.x + threadIdx.x) >> 5;
    int lane = threadIdx.x & 31;
    if (wave >= BGR * EPER) return;
    int b = wave / EPER;
    int e = wave - b * EPER;
    int s = src[b * EPER + e];
    int d = dst[b * EPER + e];
    const float4* hp = (const float4*)(h + (size_t)(b * NPER + s) * DIM);
    float4 v = hp[lane];
    float* ap = agg + (size_t)(b * NPER + d) * DIM + lane * 4;
    atomicAdd(ap + 0, v.x);
    atomicAdd(ap + 1, v.y);
    atomicAdd(ap + 2, v.z);
    atomicAdd(ap + 3, v.w);
    if (lane == 0) atomicAdd(&deg[b * NPER + d], 1.0f);
}

// agg /= max(deg, 1)
__global__ void __launch_bounds__(256)
k_scale_agg(float* __restrict__ agg, const float* __restrict__ deg) {
    int i = blockIdx.x * blockDim.x + threadIdx.x;   // float4 index
    if (i >= BGR * NPER * (DIM / 4)) return;
    int node = i / (DIM / 4);
    float inv = 1.0f / fmaxf(deg[node], 1.0f);
    float4* p = (float4*)agg;
    float4 v = p[i];
    v.x *= inv; v.y *= inv; v.z *= inv; v.w *= inv;
    p[i] = v;
}

// One wave computes z[16 nodes, 64 clusters] via fp32 WMMA, then fused
// bias + ReLU + softmax + entropy/l2 partials, stores S.
__global__ void __launch_bounds__(256)
k_gemm_softmax(const float* __restrict__ h, const float* __restrict__ agg,
               const float* __restrict__ Wself, const float* __restrict__ Wneigh,
               const float* __restrict__ bias, float* __restrict__ S,
               float* __restrict__ ent_acc, float* __restrict__ l2_acc) {
    __shared__ float lws[DIM * KDIM];
    __shared__ float lwn[DIM * KDIM];
    for (int i = threadIdx.x; i < DIM * KDIM; i += 256) {
        lws[i] = Wself[i];
        lwn[i] = Wneigh[i];
    }
    __syncthreads();

    int lane = threadIdx.x & 31;
    int tile = blockIdx.x * 8 + (threadIdx.x >> 5);  // 0..4095
    int b = tile >> 6;
    int node_base = (tile & 63) * 16;
    int half = lane >> 4;
    int l15 = lane & 15;

    v8f acc[4];
#pragma unroll
    for (int t = 0; t < 4; ++t)
#pragma unroll
        for (int r = 0; r < 8; ++r) acc[t][r] = 0.0f;

    // A-frag: lane holds row m = lane%16, k = 2*(lane/16) + j
    const float* hrow = h   + (size_t)(b * NPER + node_base + l15) * DIM;
    const float* arow = agg + (size_t)(b * NPER + node_base + l15) * DIM;

    for (int kk = 0; kk < DIM; kk += 4) {
        int k0 = kk + half * 2;
        v2f aS; aS.x = hrow[k0]; aS.y = hrow[k0 + 1];
        v2f aA; aA.x = arow[k0]; aA.y = arow[k0 + 1];
#pragma unroll
        for (int t = 0; t < 4; ++t) {
            int col = t * 16 + l15;
            v2f bS; bS.x = lws[k0 * KDIM + col]; bS.y = lws[(k0 + 1) * KDIM + col];
            v2f bN; bN.x = lwn[k0 * KDIM + col]; bN.y = lwn[(k0 + 1) * KDIM + col];
            acc[t] = __builtin_amdgcn_wmma_f32_16x16x4_f32(
                false, aS, false, bS, (short)0, acc[t], false, false);
            acc[t] = __builtin_amdgcn_wmma_f32_16x16x4_f32(
                false, aA, false, bN, (short)0, acc[t], false, false);
        }
    }

    // bias + relu   (C/D: lane col = lane%16; vgpr r -> row 8*half + r)
    float bv[4];
#pragma unroll
    for (int t = 0; t < 4; ++t) bv[t] = bias[t * 16 + l15];
#pragma unroll
    for (int t = 0; t < 4; ++t)
#pragma unroll
        for (int r = 0; r < 8; ++r)
            acc[t][r] = fmaxf(acc[t][r] + bv[t], 0.0f);

    // row softmax over 64 clusters: reduce over 4 tiles x 16 lanes (width-16)
    float entp = 0.0f, l2p = 0.0f;
#pragma unroll
    for (int r = 0; r < 8; ++r) {
        float mx = fmaxf(fmaxf(acc[0][r], acc[1][r]), fmaxf(acc[2][r], acc[3][r]));
        for (int msk = 8; msk >= 1; msk >>= 1)
            mx = fmaxf(mx, __shfl_xor(mx, msk, 16));
        float zc[4], sum = 0.0f;
#pragma unroll
        for (int t = 0; t < 4; ++t) { zc[t] = acc[t][r] - mx; sum += __expf(zc[t]); }
        for (int msk = 8; msk >= 1; msk >>= 1)
            sum += __shfl_xor(sum, msk, 16);
        float inv = 1.0f / sum;
        float lsum = __logf(sum);
        int row = node_base + half * 8 + r;
        float* Srow = S + (size_t)(b * NPER + row) * KDIM;
#pragma unroll
        for (int t = 0; t < 4; ++t) {
            float sv = __expf(zc[t]) * inv;
            entp += sv * (zc[t] - lsum);   // s * log s
            l2p  += sv * sv;
            Srow[t * 16 + l15] = sv;
        }
    }
    for (int msk = 16; msk >= 1; msk >>= 1) {
        entp += __shfl_xor(entp, msk, 32);
        l2p  += __shfl_xor(l2p,  msk, 32);
    }
    if (lane == 0) {
        atomicAdd(ent_acc, entp);
        atomicAdd(l2_acc,  l2p);
    }
}

// 1 wave per edge: AS[dst] += S[src] (float2/lane)
__global__ void __launch_bounds__(256)
k_scatter_AS(const float* __restrict__ S, const int* __restrict__ src,
             const int* __restrict__ dst, float* __restrict__ AS) {
    int wave = (blockIdx.x * blockDim.x + threadIdx.x) >> 5;
    int lane = threadIdx.x & 31;
    if (wave >= BGR * EPER) return;
    int b = wave / EPER;
    int e = wave - b * EPER;
    int s = src[b * EPER + e];
    int d = dst[b * EPER + e];
    const float2* sp = (const float2*)(S + (size_t)(b * NPER + s) * KDIM);
    float2 v = sp[lane];
    float* ap = AS + (size_t)(b * NPER + d) * KDIM + lane * 2;
    atomicAdd(ap,     v.x);
    atomicAdd(ap + 1, v.y);
}

// blocks[b] = S_b^T @ AS_b  (64x64, inner dim 1024); one wave per 16x16 tile.
__global__ void __launch_bounds__(256)
k_blocks(const float* __restrict__ S, const float* __restrict__ AS,
         float* __restrict__ out) {
    int lane = threadIdx.x & 31;
    int tile = blockIdx.x * 8 + (threadIdx.x >> 5);  // 0..1023
    int b  = tile >> 4;
    int t  = tile & 15;
    int rt = t >> 2;      // output row tile (cluster k)
    int ct = t & 3;       // output col tile (cluster m)
    int half = lane >> 4;
    int l15 = lane & 15;

    v8f acc;
#pragma unroll
    for (int r = 0; r < 8; ++r) acc[r] = 0.0f;

    const float* Sb = S  + (size_t)b * NPER * KDIM;
    const float* Ab = AS + (size_t)b * NPER * KDIM;
    int acol = rt * 16 + l15;   // A[m=k_cluster, kd=node] = S[node, k_cluster]
    int bcol = ct * 16 + l15;   // B[kd=node, n=m_cluster] = AS[node, m_cluster]

    for (int kk = 0; kk < NPER; kk += 4) {
        int k0 = kk + half * 2;
        v2f a;  a.x  = Sb[(size_t)k0 * KDIM + acol];
                a.y  = Sb[(size_t)(k0 + 1) * KDIM + acol];
        v2f bb; bb.x = Ab[(size_t)k0 * KDIM + bcol];
                bb.y = Ab[(size_t)(k0 + 1) * KDIM + bcol];
        acc = __builtin_amdgcn_wmma_f32_16x16x4_f32(
            false, a, false, bb, (short)0, acc, false, false);
    }

    int row0 = b * KDIM + rt * 16 + half * 8;
    int col  = b * KDIM + ct * 16 + l15;
#pragma unroll
    for (int r = 0; r < 8; ++r)
        out[(size_t)(row0 + r) * (BGR * KDIM) + col] = acc[r];
}

__global__ void k_finalize(const float* __restrict__ accs, float* __restrict__ out_loss) {
    if (threadIdx.x == 0 && blockIdx.x == 0) {
        float ent = -accs[0] / (float)(BGR * NPER);
        out_loss[0] = ent + sqrtf(accs[1]);
    }
}

extern "C" void kernel_launch(void* const* d_in, const int* in_sizes, int n_in,
                              void* d_out, int out_size, void* d_ws, size_t ws_size,
                              hipStream_t stream) {
    const float* h      = (const float*)d_in[0];
    const float* Wself  = (const float*)d_in[1];
    const float* Wneigh = (const float*)d_in[2];
    const float* bias   = (const float*)d_in[3];
    const int*   src    = (const int*)d_in[4];
    const int*   dst    = (const int*)d_in[5];
    float* out = (float*)d_out;
    float* ws  = (float*)d_ws;

    float* agg  = ws;                 // B*N*D
    float* deg  = ws + 8388608;       // B*N
    float* Sbuf = ws + 8454144;       // B*N*K
    float* accs = ws + 12648448;      // {ent, l2}
    float* AS   = ws;                 // reuses agg region (stream-ordered)

    // zero output (block-diagonal is sparse), agg+deg, accumulators
    hipMemsetAsync(d_out, 0, (size_t)out_size * sizeof(float), stream);
    hipMemsetAsync(agg, 0, (size_t)(8388608 + 65536) * sizeof(float), stream);
    hipMemsetAsync(accs, 0, 2 * sizeof(float), stream);

    int edge_blocks = (BGR * EPER) / 8;  // 1 wave per edge, 8 waves/block
    k_scatter_agg<<<edge_blocks, 256, 0, stream>>>(h, src, dst, agg, deg);
    k_scale_agg<<<(BGR * NPER * (DIM / 4) + 255) / 256, 256, 0, stream>>>(agg, deg);
    k_gemm_softmax<<<512, 256, 0, stream>>>(h, agg, Wself, Wneigh, bias,
                                            Sbuf, accs, accs + 1);
    hipMemsetAsync(AS, 0, (size_t)(BGR * NPER * KDIM) * sizeof(float), stream);
    k_scatter_AS<<<edge_blocks, 256, 0, stream>>>(Sbuf, src, dst, AS);
    k_blocks<<<128, 256, 0, stream>>>(Sbuf, AS, out);
    k_finalize<<<1, 1, 0, stream>>>(accs, out + (size_t)BGR * KDIM * BGR * KDIM);
}